// RealAgnosticInteractionBlock_42210938585332
// MI455X (gfx1250) — compile-verified
//
#include <hip/hip_runtime.h>
#include <hip/hip_bf16.h>
#include <math.h>

// ---------------------------------------------------------------------------
// MACE interaction block for gfx1250 (MI455X), all-f32 via V_WMMA_F32_16X16X4_F32
// ---------------------------------------------------------------------------

typedef float v2f __attribute__((ext_vector_type(2)));
typedef float v8f __attribute__((ext_vector_type(8)));

__device__ __forceinline__ v8f wmma_f32_k4(v2f a, v2f b, v8f c) {
#if __has_builtin(__builtin_amdgcn_wmma_f32_16x16x4_f32)
  // 8 args: (neg_a, A, neg_b, B, c_mod, C, reuse_a, reuse_b)
  return __builtin_amdgcn_wmma_f32_16x16x4_f32(false, a, false, b, (short)0, c,
                                               false, false);
#else
  v8f d;
  asm volatile("v_wmma_f32_16x16x4_f32 %0, %1, %2, %3"
               : "=v"(d) : "v"(a), "v"(b), "v"(c));
  return d;
#endif
}

__device__ __forceinline__ int lm_l(int m) {
  // LM_L = [0, 1,1,1, 2,2,2,2,2, 3,3,3,3,3,3,3]
  return (m == 0) ? 0 : ((m < 4) ? 1 : ((m < 9) ? 2 : 3));
}

__device__ __forceinline__ float silu(float v) {
  return v / (1.0f + __expf(-v));
}

// ---------------------------------------------------------------------------
// Zero the message accumulator.
__global__ void k_zero(float* __restrict__ p, int n) {
  int i = blockIdx.x * blockDim.x + threadIdx.x;
  if (i < n) p[i] = 0.0f;
}

// ---------------------------------------------------------------------------
// x = node_feats @ W_up * (1/sqrt(128)).  Grid: N/16 blocks, 256 threads.
// Wave w computes the 16x16 tile at cols [16w, 16w+16).
__global__ void k_upproj(const float* __restrict__ feats,
                         const float* __restrict__ Wup,
                         float* __restrict__ x) {
  const int wave = threadIdx.x >> 5;
  const int lane = threadIdx.x & 31;
  const int lo = lane & 15, hi = lane >> 4;
  const int rowBase = blockIdx.x * 16;
  const int colBase = wave * 16;
  v8f c = {};
  for (int k0 = 0; k0 < 128; k0 += 4) {
    const int kk = k0 + 2 * hi;
    const float* ap = feats + (size_t)(rowBase + lo) * 128 + kk;
    v2f a = { ap[0], ap[1] };
    v2f b = { Wup[(size_t)kk * 128 + colBase + lo],
              Wup[(size_t)(kk + 1) * 128 + colBase + lo] };
    c = wmma_f32_k4(a, b, c);
  }
  const float s = 0.08838834764831845f;  // 1/sqrt(128)
#pragma unroll
  for (int i = 0; i < 8; ++i) {
    int row = rowBase + 8 * hi + i;
    x[(size_t)row * 128 + colBase + lo] = c[i] * s;
  }
}

// ---------------------------------------------------------------------------
// Fused edge kernel: 16 edges/block. 4-layer MLP (WMMA f32) in LDS, * cutoff,
// then scatter-add mji into msg[receiver] (L2-resident accumulator).
#define N_EDGES_C 128000
__global__ void k_edge(const float* __restrict__ ef,   // edge_feats (E,8)
                       const float* __restrict__ ea,   // edge_attrs (E,16)
                       const int*   __restrict__ eidx, // (2,E)
                       const float* __restrict__ cut,  // (E,1)
                       const float* __restrict__ W1,   // (8,64)
                       const float* __restrict__ W2,   // (64,64)
                       const float* __restrict__ W3,   // (64,64)
                       const float* __restrict__ W4,   // (64,512)
                       const float* __restrict__ x,    // (N,128)
                       float* __restrict__ msg,        // (N,128,16)
                       int nEdges) {
  __shared__ float hA[16][65];
  __shared__ float hB[16][65];
  __shared__ float tp[16][516];
  __shared__ float cutS[16];

  const int t = threadIdx.x;
  const int e0 = blockIdx.x * 16;
  const int wave = t >> 5, lane = t & 31;
  const int lo = lane & 15, hi = lane >> 4;

  if (t < 128) { int el = t >> 3, k = t & 7; hA[el][k] = ef[(size_t)(e0 + el) * 8 + k]; }
  if (t < 16) cutS[t] = cut[e0 + t];
  __syncthreads();

  // Layer 1: (16x8) @ (8x64), scale 1/sqrt(8), silu.  Waves 0..3 (EXEC all-1s).
  if (wave < 4) {
    const int nt = wave;
    v8f c = {};
    for (int k0 = 0; k0 < 8; k0 += 4) {
      const int kk = k0 + 2 * hi;
      v2f a = { hA[lo][kk], hA[lo][kk + 1] };
      v2f b = { W1[kk * 64 + nt * 16 + lo], W1[(kk + 1) * 64 + nt * 16 + lo] };
      c = wmma_f32_k4(a, b, c);
    }
#pragma unroll
    for (int i = 0; i < 8; ++i)
      hB[8 * hi + i][nt * 16 + lo] = silu(c[i] * 0.35355339059327373f);
  }
  __syncthreads();

  // Layer 2: (16x64) @ (64x64), scale 1/8, silu.
  if (wave < 4) {
    const int nt = wave;
    v8f c = {};
    for (int k0 = 0; k0 < 64; k0 += 4) {
      const int kk = k0 + 2 * hi;
      v2f a = { hB[lo][kk], hB[lo][kk + 1] };
      v2f b = { W2[kk * 64 + nt * 16 + lo], W2[(kk + 1) * 64 + nt * 16 + lo] };
      c = wmma_f32_k4(a, b, c);
    }
#pragma unroll
    for (int i = 0; i < 8; ++i)
      hA[8 * hi + i][nt * 16 + lo] = silu(c[i] * 0.125f);
  }
  __syncthreads();

  // Layer 3: (16x64) @ (64x64), scale 1/8, silu.
  if (wave < 4) {
    const int nt = wave;
    v8f c = {};
    for (int k0 = 0; k0 < 64; k0 += 4) {
      const int kk = k0 + 2 * hi;
      v2f a = { hA[lo][kk], hA[lo][kk + 1] };
      v2f b = { W3[kk * 64 + nt * 16 + lo], W3[(kk + 1) * 64 + nt * 16 + lo] };
      c = wmma_f32_k4(a, b, c);
    }
#pragma unroll
    for (int i = 0; i < 8; ++i)
      hB[8 * hi + i][nt * 16 + lo] = silu(c[i] * 0.125f);
  }
  __syncthreads();

  // Layer 4: (16x64) @ (64x512), scale 1/8, * cutoff, no activation.
  // 8 waves x 4 column tiles each = 32 tiles of 16 cols.
  for (int j = 0; j < 4; ++j) {
    const int colBase = (wave * 4 + j) * 16;
    v8f c = {};
    for (int k0 = 0; k0 < 64; k0 += 4) {
      const int kk = k0 + 2 * hi;
      v2f a = { hB[lo][kk], hB[lo][kk + 1] };
      v2f b = { W4[(size_t)kk * 512 + colBase + lo],
                W4[(size_t)(kk + 1) * 512 + colBase + lo] };
      c = wmma_f32_k4(a, b, c);
    }
#pragma unroll
    for (int i = 0; i < 8; ++i) {
      int row = 8 * hi + i;
      tp[row][colBase + lo] = c[i] * 0.125f * cutS[row];
    }
  }
  __syncthreads();

  // Scatter: mji[e,u,m] = tp[e, 4u + LM_L[m]] * x[sender,u] * ea[e,m]
  // thread t -> fixed m = t&15; u = 16*i + t/16.  128 atomics/thread.
  const int m = t & 15;
  const int ug = t >> 4;
  const int l = lm_l(m);
  for (int el = 0; el < 16; ++el) {
    const int e = e0 + el;
    const int s = eidx[e];
    const int r = eidx[nEdges + e];
    const float eam = ea[(size_t)e * 16 + m];
    const float* xs = x + (size_t)s * 128;
    float* dst = msg + (size_t)r * 2048;
#pragma unroll
    for (int i = 0; i < 8; ++i) {
      const int u = i * 16 + ug;
      atomicAdd(dst + u * 16 + m, tp[el][u * 4 + l] * xs[u] * eam);
    }
  }
}

// ---------------------------------------------------------------------------
// out1[n,w,m] = (1/(16*sqrt(128))) * sum_u msg[n,u,m] * W_lin[LM_L[m]][u,w]
// Grid: (N/16)*16 blocks (Mtile, m).  Wave w -> 16-col tile.
__global__ void k_lin(const float* __restrict__ msg,
                      const float* __restrict__ Wlin,
                      float* __restrict__ out1) {
  const int m = blockIdx.x & 15;
  const int mtile = blockIdx.x >> 4;
  const int wave = threadIdx.x >> 5, lane = threadIdx.x & 31;
  const int lo = lane & 15, hi = lane >> 4;
  const float* W = Wlin + (size_t)lm_l(m) * 128 * 128;
  const int rowBase = mtile * 16, colBase = wave * 16;
  v8f c = {};
  for (int k0 = 0; k0 < 128; k0 += 4) {
    const int kk = k0 + 2 * hi;
    const float* mp = msg + (size_t)(rowBase + lo) * 2048 + (size_t)kk * 16 + m;
    v2f a = { mp[0], mp[16] };
    v2f b = { W[(size_t)kk * 128 + colBase + lo],
              W[(size_t)(kk + 1) * 128 + colBase + lo] };
    c = wmma_f32_k4(a, b, c);
  }
  const float s = 0.08838834764831845f / 16.0f;  // (1/sqrt(128)) / AVG_NUM_NEIGHBORS
#pragma unroll
  for (int i = 0; i < 8; ++i) {
    int row = rowBase + 8 * hi + i;
    out1[(size_t)row * 2048 + (size_t)(colBase + lo) * 16 + m] = c[i] * s;
  }
}

// ---------------------------------------------------------------------------
// Skip contraction, one node per block:
//   out[n,w,m] = norm * sum_{u,v} out1[n,u,m] * attrs[n,v] * W_skip[l][u,v,w]
// Build A[u,w] = sum_v attrs[v]*W_skip[l][u,v,w] in LDS (two 64-u halves),
// apply to the node's cached block.  Reads and overwrites d_out (own node only).
__global__ void k_skip(const float* __restrict__ out1,
                       const float* __restrict__ attrs,
                       const float* __restrict__ Wskip,
                       float* __restrict__ out) {
  __shared__ float As[64 * 128];   // 32 KB
  __shared__ float blk[2048];      //  8 KB
  __shared__ float av[10];
  const int n = blockIdx.x;
  const int t = threadIdx.x;
  const int w = t & 127, mrow = t >> 7;

  if (t < 10) av[t] = attrs[(size_t)n * 10 + t];
#pragma unroll
  for (int i = 0; i < 8; ++i)
    blk[t + i * 256] = out1[(size_t)n * 2048 + t + i * 256];
  __syncthreads();

  const int LOFF[5] = {0, 1, 4, 9, 16};
  const float norm = 0.027950849718747371f;  // 1/sqrt(1280)

  for (int l = 0; l < 4; ++l) {
    const int dim = 2 * l + 1;
    float acc[4] = {0.f, 0.f, 0.f, 0.f};
    const float* Wl = Wskip + (size_t)l * 128 * 10 * 128;
    for (int uh = 0; uh < 128; uh += 64) {
      __syncthreads();  // protect As reuse
      for (int i = 0; i < 32; ++i) {
        const int e = t + i * 256;
        const int ul = e >> 7, ww = e & 127;
        const float* wp = Wl + (size_t)(uh + ul) * 1280 + ww;
        float s = 0.f;
#pragma unroll
        for (int v = 0; v < 10; ++v) s += av[v] * wp[v * 128];
        As[e] = s;
      }
      __syncthreads();
      for (int ul = 0; ul < 64; ++ul) {
        const float Aval = As[ul * 128 + w];
        const float* bp = blk + (uh + ul) * 16 + LOFF[l];
        int jj = 0;
        for (int j = mrow; j < dim; j += 2, ++jj) acc[jj] += bp[j] * Aval;
      }
    }
    __syncthreads();  // done reading blk/As for this l before next overwrite
    int jj = 0;
    for (int j = mrow; j < dim; j += 2, ++jj)
      out[(size_t)n * 2048 + w * 16 + LOFF[l] + j] = acc[jj] * norm;
  }
}

// ---------------------------------------------------------------------------
extern "C" void kernel_launch(void* const* d_in, const int* in_sizes, int n_in,
                              void* d_out, int out_size, void* d_ws, size_t ws_size,
                              hipStream_t stream) {
  const float* node_attrs = (const float*)d_in[0];
  const float* node_feats = (const float*)d_in[1];
  const float* edge_attrs = (const float*)d_in[2];
  const float* edge_feats = (const float*)d_in[3];
  const int*   edge_index = (const int*)d_in[4];
  const float* cutoff     = (const float*)d_in[5];
  const float* W_up   = (const float*)d_in[6];
  const float* W_mlp1 = (const float*)d_in[7];
  const float* W_mlp2 = (const float*)d_in[8];
  const float* W_mlp3 = (const float*)d_in[9];
  const float* W_mlp4 = (const float*)d_in[10];
  const float* W_lin  = (const float*)d_in[11];
  const float* W_skip = (const float*)d_in[12];

  const int N = in_sizes[1] / 128;   // 8000 nodes
  const int E = in_sizes[3] / 8;     // 128000 edges

  float* out = (float*)d_out;
  float* x   = (float*)d_ws;                 // N*128 floats  (4.1 MB)
  float* msg = x + (size_t)N * 128;          // N*2048 floats (65.5 MB)

  const int msgN = N * 2048;
  k_zero<<<(msgN + 255) / 256, 256, 0, stream>>>(msg, msgN);
  k_upproj<<<N / 16, 256, 0, stream>>>(node_feats, W_up, x);
  k_edge<<<E / 16, 256, 0, stream>>>(edge_feats, edge_attrs, edge_index, cutoff,
                                     W_mlp1, W_mlp2, W_mlp3, W_mlp4, x, msg, E);
  k_lin<<<(N / 16) * 16, 256, 0, stream>>>(msg, W_lin, out);
  k_skip<<<N, 256, 0, stream>>>(out, node_attrs, W_skip, out);
}